// HorseConv_46892452938214
// MI455X (gfx1250) — compile-verified
//
#include <hip/hip_runtime.h>
#include <stdint.h>

typedef __attribute__((ext_vector_type(2))) float v2f;
typedef __attribute__((ext_vector_type(8))) float v8f;

#define B_      32
#define CIN     64
#define COUT    64
#define H_      128
#define W_      128
#define HOUT    124
#define WOUT    124
#define TILE_M  64
#define LDSTR   68                 // padded LDS row stride (floats): b64 reads hit all 64 banks
#define SLAB    (64 * LDSTR)       // 4352 floats = 17408 bytes per slab

// One async 4B global->LDS copy per lane. GVS addressing:
//   mem = SADDR(64b sgpr pair) + VADDR(32b vgpr) ; VDST holds LDS byte address.
__device__ __forceinline__ void async_g2l_b32(uint32_t lds_off, uint32_t goff,
                                              uint64_t sbase) {
    asm volatile("global_load_async_to_lds_b32 %0, %1, %2"
                 :: "v"(lds_off), "v"(goff), "s"(sbase) : "memory");
}

__global__ __launch_bounds__(256)
void HorseConv_wmma_async(const float* __restrict__ x,
                          const float* __restrict__ w,     // (9, Cout, Cin)
                          const float* __restrict__ bias,  // (Cout)
                          float* __restrict__ out) {
    // [ X buffer 0 | X buffer 1 | W buffer ]  = 52224 bytes
    // X buffers: Xs[pixel][cin]  (M-major -> A-frag is one ds_load_b64)
    // W buffer : Ws[cout][cin]   (N-major -> B-frag is one ds_load_b64, no transpose)
    __shared__ float smem[3 * SLAB];

    const int tid  = threadIdx.x;
    const int lane = tid & 31;
    const int wave = tid >> 5;

    const int b   = blockIdx.z;
    const int oy  = blockIdx.y;
    const int ox0 = (blockIdx.x == 0) ? 0 : (WOUT - TILE_M);  // 0 or 60, overlap -> no OOB

    // 8 waves tile the 64x64 (M x N) output: 4 M-rows x 2 N-cols of 16x32 pieces
    const int m_base = (wave >> 1) * 16;
    const int n_base = (wave & 1) * 32;

    // WMMA f32 16x16x4 operand layout (wave32):
    //  A (16x4): lanes 0-15 hold K=0(v0),K=1(v1); lanes 16-31 hold K=2,K=3; M = lane&15
    //  B (4x16): mirrored; N = lane&15
    const int am = lane & 15;
    const int kh = (lane >> 4) * 2;

    v8f acc0 = {};
    v8f acc1 = {};

    // knight-move taps + center: HORSE_POS = {1,3,5,9,12,15,19,21,23} in a 5x5
    const int DY[9] = {0, 0, 1, 1, 2, 3, 3, 4, 4};
    const int DX[9] = {1, 3, 0, 4, 2, 0, 4, 1, 3};

    const uint64_t xbase = (uint64_t)(uintptr_t)x +
                           (uint64_t)b * (uint64_t)(CIN * H_ * W_) * 4u;
    const uint64_t wbase = (uint64_t)(uintptr_t)w;

    // Staging decomposition: element i = tid + j*256 -> c = (tid>>6) + 4*j, p = tid&63
    const uint32_t c0 = (uint32_t)(tid >> 6);
    const uint32_t p  = (uint32_t)(tid & 63);

    // ---- prologue: async-stage x slab for tap 0 into X buffer 0 (Xs[p][c]) ----
    {
        uint32_t go = ((uint32_t)((oy + DY[0]) * W_ + ox0 + DX[0]) +
                       c0 * (uint32_t)(H_ * W_) + p) * 4u;
        uint32_t lo = (p * LDSTR + c0) * 4u;
        #pragma unroll
        for (int j = 0; j < 16; ++j) {
            async_g2l_b32(lo, go, xbase);
            go += 4u * (uint32_t)(H_ * W_) * 4u;   // c += 4 channels
            lo += 4u * 4u;
        }
    }

    for (int tap = 0; tap < 9; ++tap) {
        const uint32_t cur = (uint32_t)(tap & 1);

        // Stage this tap's weights: Ws[cout][cin] = w[tap,cout,cin] (natural layout).
        // Previous tap's Ws readers finished at the trailing barrier below.
        {
            uint32_t go = ((uint32_t)tap * (uint32_t)(COUT * CIN) +
                           c0 * (uint32_t)CIN + p) * 4u;            // co=c0+4j, ci=p
            uint32_t lo = (2u * SLAB + c0 * LDSTR + p) * 4u;        // Ws[co][ci]
            #pragma unroll
            for (int j = 0; j < 16; ++j) {
                async_g2l_b32(lo, go, wbase);
                go += 4u * (uint32_t)CIN * 4u;                      // co += 4
                lo += 4u * LDSTR * 4u;
            }
        }

        if (tap + 1 < 9) {
            // Prefetch next tap's x slab into the other X buffer, then wait for the
            // OLDEST 32 (X(t) + W(t)) only: async loads complete in order, so
            // asynccnt <= 16 leaves just the 16 newest (X(t+1)) in flight under the MMAs.
            uint32_t go = ((uint32_t)((oy + DY[tap + 1]) * W_ + ox0 + DX[tap + 1]) +
                           c0 * (uint32_t)(H_ * W_) + p) * 4u;
            uint32_t lo = ((1u - cur) * SLAB + p * LDSTR + c0) * 4u;
            #pragma unroll
            for (int j = 0; j < 16; ++j) {
                async_g2l_b32(lo, go, xbase);
                go += 4u * (uint32_t)(H_ * W_) * 4u;
                lo += 4u * 4u;
            }
            asm volatile("s_wait_asynccnt 0x10" ::: "memory");
        } else {
            asm volatile("s_wait_asynccnt 0x0" ::: "memory");
        }
        __syncthreads();   // publish X(t), W(t) LDS writes to all waves

        const float* Xc = smem + cur * SLAB;
        const float* Wp = smem + 2 * SLAB;

        #pragma unroll
        for (int k0 = 0; k0 < CIN; k0 += 4) {
            // One 8B LDS load per fragment: lane's (k, k+1) pair is contiguous.
            v2f a  = *(const v2f*)&Xc[(m_base + am)      * LDSTR + k0 + kh];
            v2f b0 = *(const v2f*)&Wp[(n_base + am)      * LDSTR + k0 + kh];
            v2f b1 = *(const v2f*)&Wp[(n_base + 16 + am) * LDSTR + k0 + kh];
            acc0 = __builtin_amdgcn_wmma_f32_16x16x4_f32(
                false, a, false, b0, (short)0, acc0, false, false);
            acc1 = __builtin_amdgcn_wmma_f32_16x16x4_f32(
                false, a, false, b1, (short)0, acc1, false, false);
        }
        __syncthreads();   // all waves done reading Ws before next tap overwrites it
    }

    // Stage D tiles through LDS (reuse X buffer 0 as staged[cout][pixel]) so the
    // global stores are coalesced per Cout row.
    // D layout: VGPR r -> M = m_base + (lane>=16 ? 8 : 0) + r, N = n_base(+16) + (lane&15)
    {
        float* St = smem;
        const int mrow = m_base + (lane >> 4) * 8;
        #pragma unroll
        for (int r = 0; r < 8; ++r) {
            St[(n_base      + am) * LDSTR + mrow + r] = acc0[r];
            St[(n_base + 16 + am) * LDSTR + mrow + r] = acc1[r];
        }
    }
    __syncthreads();
    {
        const float* St = smem;
        float* dst = out + (long)b * (COUT * HOUT * WOUT) + (long)oy * WOUT + ox0;
        for (int i = tid; i < COUT * TILE_M; i += 256) {
            const int n = i >> 6;
            const int m = i & 63;
            dst[(long)n * (HOUT * WOUT) + m] = St[n * LDSTR + m] + bias[n];
        }
    }
}

extern "C" void kernel_launch(void* const* d_in, const int* in_sizes, int n_in,
                              void* d_out, int out_size, void* d_ws, size_t ws_size,
                              hipStream_t stream) {
    (void)in_sizes; (void)n_in; (void)d_ws; (void)ws_size; (void)out_size;
    const float* x    = (const float*)d_in[0];
    const float* w    = (const float*)d_in[1];
    const float* bias = (const float*)d_in[2];
    float* out        = (float*)d_out;

    dim3 grid(2, HOUT, B_);   // (ox tiles, output rows, batch)
    dim3 block(256);
    HorseConv_wmma_async<<<grid, block, 0, stream>>>(x, w, bias, out);
}